// RNN_28432683500094
// MI455X (gfx1250) — compile-verified
//
#include <hip/hip_runtime.h>

typedef float v2f __attribute__((ext_vector_type(2)));
typedef float v8f __attribute__((ext_vector_type(8)));

#define Tdim 512
#define Bdim 64
#define Hdim 1024              // == I, so x rows and h rows have identical stride
#define BH   (Bdim * Hdim)
#define NTILES 256             // (Bdim/16) * (Hdim/16) for the generic kernel
#define NSTRIPS 64             // Hdim/16 N-strips for the register kernel

// ---------------- device-wide split barrier (generation based) ----------------
__device__ __forceinline__ void grid_barrier(unsigned* cnt, unsigned* gen) {
    __syncthreads();
    if (threadIdx.x == 0) {
        const unsigned g = __atomic_load_n(gen, __ATOMIC_RELAXED);
        __threadfence();                                   // release my writes
        if (atomicAdd(cnt, 1u) == gridDim.x - 1) {
            __atomic_store_n(cnt, 0u, __ATOMIC_RELAXED);   // reset for next phase
            __threadfence();
            __atomic_store_n(gen, g + 1u, __ATOMIC_RELAXED);
            __threadfence();                               // acquire others' writes
        } else {
            while (__atomic_load_n(gen, __ATOMIC_RELAXED) == g) {
                __builtin_amdgcn_s_sleep(8);
            }
            __threadfence();                               // acquire others' writes
        }
    }
    __syncthreads();
}

__global__ void init_sync(unsigned* p) { p[0] = 0u; p[16] = 0u; }

// =============================================================================
// Register-resident-weights persistent kernel (preferred path).
// Grid: 64 blocks, one 16-wide N-strip each; 8 wave32s split K=2048
// (waves 0-3: x@w_ih quarter, waves 4-7: h@w_hh quarter).  Each wave holds its
// whole B slice (64 x v2f = 128 VGPRs) in registers for all 512 steps and
// computes 4 m-tiles (B=64 rows) so one B frag feeds 4 WMMAs per k-step.
// =============================================================================
__global__ __launch_bounds__(256, 1) void rnn_reg(
    const float* xseq,                     // [T,B,1024] step inputs
    float*       outseq,                   // output stream (may alias xseq, L1)
    const float* __restrict__ h0,          // hx[layer]
    float* __restrict__ hbuf0,             // ping  [B,H]
    float* __restrict__ hbuf1,             // pong  [B,H]
    float* __restrict__ hn_dst,            // h_n slot for this layer
    const float* __restrict__ wx,
    const float* __restrict__ wh,
    const float* __restrict__ bih,
    const float* __restrict__ bhh,
    unsigned* bar_cnt, unsigned* bar_gen)
{
    __shared__ float red[4][8][8][32];     // [m-tile][src wave][C vgpr][lane]

    const int tid     = threadIdx.x;
    const int wave    = tid >> 5;
    const int lane    = tid & 31;
    const int lane_lo = lane & 15;
    const int lane_hi = lane >> 4;
    const int n0      = blockIdx.x << 4;   // this block's hidden-col strip
    const int col     = n0 + lane_lo;

    const int part  = wave >> 2;           // 0: x-part, 1: h-part
    const int kbase = (wave & 3) << 8;     // 0,256,512,768
    const int koff  = kbase + (lane_hi << 1);

    // ---- Preload this wave's weight slice into registers (invariant over t).
    // B frag (4x16 f32): lanes 0-15 K={k,k+1}, lanes 16-31 K={k+2,k+3}.
    const float* wcol = (part ? wh : wx) + (size_t)koff * Hdim + col;
    v2f breg[64];
    #pragma unroll
    for (int i = 0; i < 64; ++i) {
        breg[i].x = wcol[(size_t)(4 * i) * Hdim];
        breg[i].y = wcol[(size_t)(4 * i + 1) * Hdim];
    }
    const float bsum = bih[col] + bhh[col];

    for (int t = 0; t < Tdim; ++t) {
        const float* hprev = (t == 0) ? h0 : ((t & 1) ? hbuf0 : hbuf1);
        float*       hcur  = (t & 1) ? hbuf1 : hbuf0;
        const float* xin   = xseq + (size_t)t * BH;

        // Stream h_{t-1} strip into output slot t-1 (nobody reads slot t-1).
        if (t > 0) {
            float* prev_store = outseq + (size_t)(t - 1) * BH;
            #pragma unroll
            for (int j = 0; j < 4; ++j) {
                const int r = (tid >> 4) + (j << 4);
                const int c = n0 + (tid & 15);
                prev_store[r * Hdim + c] = hprev[r * Hdim + c];
            }
        }

        // A frag (16x4 f32): 2 consecutive K floats per lane; 4 m-tiles
        // share each register-resident B frag.
        const float* ab = (part ? hprev : xin) + (size_t)lane_lo * Hdim + koff;

        v8f acc0 = {}, acc1 = {}, acc2 = {}, acc3 = {};
        #pragma unroll
        for (int i = 0; i < 64; ++i) {
            const float* ap = ab + 4 * i;
            v2f a0 = *(const v2f*)(ap);
            v2f a1 = *(const v2f*)(ap + 16 * Hdim);
            v2f a2 = *(const v2f*)(ap + 32 * Hdim);
            v2f a3 = *(const v2f*)(ap + 48 * Hdim);
            acc0 = __builtin_amdgcn_wmma_f32_16x16x4_f32(false, a0, false, breg[i], (short)0, acc0, false, false);
            acc1 = __builtin_amdgcn_wmma_f32_16x16x4_f32(false, a1, false, breg[i], (short)0, acc1, false, false);
            acc2 = __builtin_amdgcn_wmma_f32_16x16x4_f32(false, a2, false, breg[i], (short)0, acc2, false, false);
            acc3 = __builtin_amdgcn_wmma_f32_16x16x4_f32(false, a3, false, breg[i], (short)0, acc3, false, false);
        }

        #pragma unroll
        for (int i = 0; i < 8; ++i) {
            red[0][wave][i][lane] = acc0[i];
            red[1][wave][i][lane] = acc1[i];
            red[2][wave][i][lane] = acc2[i];
            red[3][wave][i][lane] = acc3[i];
        }
        __syncthreads();

        // Wave w owns C-VGPR index w: row = 16*m + w + 8*lane_hi
        #pragma unroll
        for (int m = 0; m < 4; ++m) {
            float s = 0.0f;
            #pragma unroll
            for (int src = 0; src < 8; ++src) s += red[m][src][wave][lane];
            s += bsum;
            s = fmaxf(s, 0.0f);
            const int row = (m << 4) + wave + (lane_hi << 3);
            hcur[(size_t)row * Hdim + col] = s;
        }

        grid_barrier(bar_cnt, bar_gen);    // also fences LDS reuse across steps
    }

    // hlast -> output slot T-1 and h_n (after last barrier: safe even when
    // outseq aliases xseq in layer 1).
    const float* hlast = (Tdim & 1) ? hbuf1 : hbuf0;   // == hbuf0, last written at step T-2
    #pragma unroll
    for (int j = 0; j < 4; ++j) {
        const int r = (tid >> 4) + (j << 4);
        const int c = n0 + (tid & 15);
        const float v = hlast[r * Hdim + c];
        outseq[(size_t)(Tdim - 1) * BH + r * Hdim + c] = v;
        hn_dst[r * Hdim + c] = v;
    }
}

// =============================================================================
// Generic fallback: tile-striding persistent kernel (any co-resident grid).
// =============================================================================
__global__ __launch_bounds__(256) void rnn_generic(
    const float* xseq,
    float*       outseq,
    const float* __restrict__ h0,
    float* __restrict__ hbuf0,
    float* __restrict__ hbuf1,
    float* __restrict__ hn_dst,
    const float* __restrict__ wx,
    const float* __restrict__ wh,
    const float* __restrict__ bih,
    const float* __restrict__ bhh,
    unsigned* bar_cnt, unsigned* bar_gen)
{
    __shared__ float red[8][8][32];

    const int tid     = threadIdx.x;
    const int wave    = tid >> 5;
    const int lane    = tid & 31;
    const int lane_lo = lane & 15;
    const int lane_hi = lane >> 4;

    for (int t = 0; t < Tdim; ++t) {
        const float* hprev = (t == 0) ? h0 : ((t & 1) ? hbuf0 : hbuf1);
        float*       hcur  = (t & 1) ? hbuf1 : hbuf0;
        const float* xin   = xseq + (size_t)t * BH;
        float*       prev_store = (t > 0) ? outseq + (size_t)(t - 1) * BH : nullptr;

        for (int tile = blockIdx.x; tile < NTILES; tile += gridDim.x) {
            const int m0 = (tile >> 6) << 4;
            const int n0 = (tile & 63) << 4;

            if (prev_store != nullptr) {
                const int r = m0 + (tid >> 4);
                const int c = n0 + (tid & 15);
                prev_store[r * Hdim + c] = hprev[r * Hdim + c];
            }

            const float* Amat = (wave & 4) ? hprev : xin;
            const float* Wmat = (wave & 4) ? wh    : wx;
            const int kbase   = (wave & 3) << 8;

            const float* arow = Amat + (size_t)(m0 + lane_lo) * Hdim
                                     + kbase + (lane_hi << 1);
            const float* wcol = Wmat + (size_t)(kbase + (lane_hi << 1)) * Hdim
                                     + n0 + lane_lo;

            v8f acc = {};
            #pragma unroll 8
            for (int kk = 0; kk < 256; kk += 4) {
                v2f a = *(const v2f*)(arow + kk);
                v2f b;
                b.x = wcol[(size_t)kk * Hdim];
                b.y = wcol[(size_t)(kk + 1) * Hdim];
                acc = __builtin_amdgcn_wmma_f32_16x16x4_f32(
                          false, a, false, b, (short)0, acc, false, false);
            }

            #pragma unroll
            for (int i = 0; i < 8; ++i) red[wave][i][lane] = acc[i];
            __syncthreads();

            float s = 0.0f;
            #pragma unroll
            for (int src = 0; src < 8; ++src) s += red[src][wave][lane];

            const int col = n0 + lane_lo;
            s += bih[col] + bhh[col];
            s = fmaxf(s, 0.0f);
            const int row = m0 + wave + (lane_hi << 3);
            hcur[(size_t)row * Hdim + col] = s;

            __syncthreads();
        }

        grid_barrier(bar_cnt, bar_gen);
    }

    const float* hlast = (Tdim & 1) ? hbuf1 : hbuf0;
    for (int tile = blockIdx.x; tile < NTILES; tile += gridDim.x) {
        const int m0 = (tile >> 6) << 4;
        const int n0 = (tile & 63) << 4;
        const int r  = m0 + (tid >> 4);
        const int c  = n0 + (tid & 15);
        const float v = hlast[r * Hdim + c];
        outseq[(size_t)(Tdim - 1) * BH + r * Hdim + c] = v;
        hn_dst[r * Hdim + c] = v;
    }
}

extern "C" void kernel_launch(void* const* d_in, const int* in_sizes, int n_in,
                              void* d_out, int out_size, void* d_ws, size_t ws_size,
                              hipStream_t stream) {
    const float* x     = (const float*)d_in[0];
    const float* hx    = (const float*)d_in[1];
    const float* w_ih0 = (const float*)d_in[2];
    const float* w_hh0 = (const float*)d_in[3];
    const float* b_ih0 = (const float*)d_in[4];
    const float* b_hh0 = (const float*)d_in[5];
    const float* w_ih1 = (const float*)d_in[6];
    const float* w_hh1 = (const float*)d_in[7];
    const float* b_ih1 = (const float*)d_in[8];
    const float* b_hh1 = (const float*)d_in[9];

    float* out = (float*)d_out;                  // out1 [T,B,H] ++ h_n [2,B,H]

    float*    hb0  = (float*)d_ws;               // ping  [B,H]
    float*    hb1  = hb0 + BH;                   // pong  [B,H]
    unsigned* sync = (unsigned*)(hb1 + BH);      // [0]=count, [16]=generation

    // Pick persistent config; all blocks must be co-resident.
    int mp = 0;
    {
        int dev = 0;
        hipDeviceProp_t prop;
        if (hipGetDevice(&dev) == hipSuccess &&
            hipGetDeviceProperties(&prop, dev) == hipSuccess)
            mp = prop.multiProcessorCount;
    }
    int occ_reg = 0, occ_gen = 0;
    (void)hipOccupancyMaxActiveBlocksPerMultiprocessor(&occ_reg, rnn_reg, 256, 0);
    (void)hipOccupancyMaxActiveBlocksPerMultiprocessor(&occ_gen, rnn_generic, 256, 0);

    const bool use_reg = (mp > 0) && (occ_reg > 0) &&
                         ((long long)occ_reg * mp >= NSTRIPS);
    int nblocks_gen = 16;                         // conservative fallback
    if (mp > 0 && occ_gen > 0) {
        long long nb = (long long)occ_gen * mp;
        nblocks_gen = (int)(nb < 1 ? 1 : (nb > NTILES ? NTILES : nb));
    }

    init_sync<<<1, 1, 0, stream>>>(sync);

    if (use_reg) {
        rnn_reg<<<NSTRIPS, 256, 0, stream>>>(
            x, out, hx, hb0, hb1, out + (size_t)Tdim * BH,
            w_ih0, w_hh0, b_ih0, b_hh0, sync, sync + 16);
        rnn_reg<<<NSTRIPS, 256, 0, stream>>>(
            out, out, hx + BH, hb0, hb1, out + (size_t)(Tdim + 1) * BH,
            w_ih1, w_hh1, b_ih1, b_hh1, sync, sync + 16);
    } else {
        rnn_generic<<<nblocks_gen, 256, 0, stream>>>(
            x, out, hx, hb0, hb1, out + (size_t)Tdim * BH,
            w_ih0, w_hh0, b_ih0, b_hh0, sync, sync + 16);
        rnn_generic<<<nblocks_gen, 256, 0, stream>>>(
            out, out, hx + BH, hb0, hb1, out + (size_t)(Tdim + 1) * BH,
            w_ih1, w_hh1, b_ih1, b_hh1, sync, sync + 16);
    }
}